// GraphContextEncoder_11768210391412
// MI455X (gfx1250) — compile-verified
//
#include <hip/hip_runtime.h>
#include <math.h>
#include <float.h>

#define NN   20000
#define EE   160000
#define GG   64
#define DIN  512
#define HIDC 256
#define OUTC 256
#define NH1  4
#define EED  16
#define C1   1024   // NH1*HIDC

typedef float v2f __attribute__((ext_vector_type(2)));
typedef float v8f __attribute__((ext_vector_type(8)));

__device__ __forceinline__ float wave_sum(float v) {
#pragma unroll
  for (int o = 16; o > 0; o >>= 1) v += __shfl_xor(v, o, 32);
  return v;
}

__global__ void fill_f32(float* __restrict__ p, float v, int n) {
  int i = blockIdx.x * blockDim.x + threadIdx.x;
  if (i < n) p[i] = v;
}

// e1 = relu(edge_attr @ We_a + be_a) @ We_b + be_b          [E,16]
__global__ void edge_encoder(const float* __restrict__ ea,
                             const float* __restrict__ Wa, const float* __restrict__ ba,
                             const float* __restrict__ Wb, const float* __restrict__ bb,
                             float* __restrict__ e1) {
  int e = blockIdx.x * blockDim.x + threadIdx.x;
  if (e >= EE) return;
  float v = ea[e];
  float t[EED];
#pragma unroll
  for (int j = 0; j < EED; j++) {
    float u = fmaf(v, Wa[j], ba[j]);
    t[j] = u > 0.f ? u : 0.f;
  }
#pragma unroll
  for (int k = 0; k < EED; k++) {
    float s = bb[k];
#pragma unroll
    for (int j = 0; j < EED; j++) s = fmaf(t[j], Wb[j * EED + k], s);
    e1[(size_t)e * EED + k] = s;
  }
}

// Y[M,Nout] = X[M,K] @ W[K,Nout] + bias, f32 WMMA 16x16x4.
// One wave computes a 16(M) x 64(N) strip: A tile reused across 4 WMMAs/K-step.
__global__ void gemm_wmma(const float* __restrict__ X, const float* __restrict__ W,
                          const float* __restrict__ bias, float* __restrict__ Y,
                          int M, int K, int Nout) {
  int wid  = blockIdx.x * (blockDim.x >> 5) + (threadIdx.x >> 5);
  int lane = threadIdx.x & 31;
  int tilesN = Nout >> 6;
  int tm = wid / tilesN;
  int tn = wid - tm * tilesN;
  if (tm >= (M >> 4)) return;              // wave-uniform: EXEC stays all-ones
  int m0 = tm << 4, n0 = tn << 6;
  int half = lane >> 4;                    // K-pair selector (ISA A/B layout)
  int l15  = lane & 15;

  v8f acc0 = {}, acc1 = {}, acc2 = {}, acc3 = {};
  const float* xrow = X + (size_t)(m0 + l15) * K;

  for (int k = 0; k < K; k += 4) {
    int kk = k + half * 2;
    v2f a = *(const v2f*)(xrow + kk);      // A[m, kk], A[m, kk+1]
    const float* w0 = W + (size_t)kk * Nout + n0 + l15;
    const float* w1 = w0 + Nout;
    v2f b0; b0.x = w0[0];  b0.y = w1[0];
    v2f b1; b1.x = w0[16]; b1.y = w1[16];
    v2f b2; b2.x = w0[32]; b2.y = w1[32];
    v2f b3; b3.x = w0[48]; b3.y = w1[48];
    acc0 = __builtin_amdgcn_wmma_f32_16x16x4_f32(false, a, false, b0, (short)0, acc0, false, false);
    acc1 = __builtin_amdgcn_wmma_f32_16x16x4_f32(false, a, false, b1, (short)0, acc1, false, false);
    acc2 = __builtin_amdgcn_wmma_f32_16x16x4_f32(false, a, false, b2, (short)0, acc2, false, false);
    acc3 = __builtin_amdgcn_wmma_f32_16x16x4_f32(false, a, false, b3, (short)0, acc3, false, false);
  }

  float bv0 = bias[n0 + l15];
  float bv1 = bias[n0 + l15 + 16];
  float bv2 = bias[n0 + l15 + 32];
  float bv3 = bias[n0 + l15 + 48];
  int mbase = m0 + (half << 3);
#pragma unroll
  for (int r = 0; r < 8; r++) {
    float* yrow = Y + (size_t)(mbase + r) * Nout + n0 + l15;
    yrow[0]  = acc0[r] + bv0;
    yrow[16] = acc1[r] + bv1;
    yrow[32] = acc2[r] + bv2;
    yrow[48] = acc3[r] + bv3;
  }
}

// conv1 attention logits: one wave per edge; Wed1 staged in 64KB LDS.
__global__ void edge_logits1(const float* __restrict__ xl, const float* __restrict__ xr,
                             const float* __restrict__ e1, const float* __restrict__ Wed,
                             const float* __restrict__ att, const int* __restrict__ ei,
                             float* __restrict__ logits, float* __restrict__ nmax) {
  __shared__ float sW[EED * C1];           // 64 KB
  for (int i = threadIdx.x; i < EED * C1; i += blockDim.x) sW[i] = Wed[i];
  __syncthreads();
  int wid  = blockIdx.x * 8 + (threadIdx.x >> 5);
  int lane = threadIdx.x & 31;
  if (wid >= EE) return;
  int s = ei[wid], d = ei[EE + wid];
  const float* xld = xl + (size_t)d * C1;
  const float* xrs = xr + (size_t)s * C1;
  float ev[EED];
#pragma unroll
  for (int k = 0; k < EED; k++) ev[k] = e1[(size_t)wid * EED + k];
#pragma unroll
  for (int h = 0; h < NH1; h++) {
    float acc = 0.f;
#pragma unroll
    for (int jj = 0; jj < 8; jj++) {
      int c = h * 256 + jj * 32 + lane;
      float ep = 0.f;
#pragma unroll
      for (int k = 0; k < EED; k++) ep = fmaf(ev[k], sW[k * C1 + c], ep);
      float m = xld[c] + xrs[c] + ep;
      m = m > 0.f ? m : 0.2f * m;          // leaky_relu(0.2)
      acc = fmaf(att[c], m, acc);          // att1 flat [4*256] == index c
    }
    acc = wave_sum(acc);
    if (lane == 0) {
      logits[(size_t)wid * NH1 + h] = acc;
      atomicMax(&nmax[d * NH1 + h], acc);
    }
  }
}

__global__ void edge_exp1(float* __restrict__ a, const float* __restrict__ nmax,
                          float* __restrict__ nden, const int* __restrict__ ei) {
  int i = blockIdx.x * blockDim.x + threadIdx.x;
  if (i >= EE * NH1) return;
  int e = i >> 2, h = i & 3;
  int d = ei[EE + e];
  float x = expf(a[i] - nmax[d * NH1 + h]);
  a[i] = x;
  atomicAdd(&nden[d * NH1 + h], x);
}

__global__ void edge_aggr1(const float* __restrict__ xr, const float* __restrict__ a,
                           const float* __restrict__ nden, const int* __restrict__ ei,
                           float* __restrict__ out) {
  int wid  = blockIdx.x * 8 + (threadIdx.x >> 5);
  int lane = threadIdx.x & 31;
  if (wid >= EE) return;
  int s = ei[wid], d = ei[EE + wid];
  const float* xrs = xr + (size_t)s * C1;
  float* od = out + (size_t)d * C1;
#pragma unroll
  for (int h = 0; h < NH1; h++) {
    float alpha = a[(size_t)wid * NH1 + h] / (nden[d * NH1 + h] + 1e-16f);
#pragma unroll
    for (int jj = 0; jj < 8; jj++) {
      int c = h * 256 + jj * 32 + lane;
      atomicAdd(&od[c], xrs[c] * alpha);
    }
  }
}

// h = gelu(layernorm(out1 + bias1)) in place; one wave per node row of 1024.
__global__ void ln_gelu1(float* __restrict__ h, const float* __restrict__ bias,
                         const float* __restrict__ g, const float* __restrict__ beta) {
  int n = blockIdx.x * 8 + (threadIdx.x >> 5);
  int lane = threadIdx.x & 31;
  if (n >= NN) return;
  float* row = h + (size_t)n * C1;
  float v[32];
  float s = 0.f;
#pragma unroll
  for (int j = 0; j < 32; j++) { int c = j * 32 + lane; v[j] = row[c] + bias[c]; s += v[j]; }
  float mu = wave_sum(s) * (1.f / C1);
  float q = 0.f;
#pragma unroll
  for (int j = 0; j < 32; j++) { float dd = v[j] - mu; q += dd * dd; }
  float var = wave_sum(q) * (1.f / C1);
  float rs = rsqrtf(var + 1e-5f);
#pragma unroll
  for (int j = 0; j < 32; j++) {
    int c = j * 32 + lane;
    float t = (v[j] - mu) * rs * g[c] + beta[c];
    row[c] = 0.5f * t * (1.f + erff(t * 0.70710678118f));   // exact GELU
  }
}

// conv2 logits (heads=1, C=256): one wave per edge; Wed2 in 16KB LDS.
__global__ void edge_logits2(const float* __restrict__ xl, const float* __restrict__ xr,
                             const float* __restrict__ e1, const float* __restrict__ Wed,
                             const float* __restrict__ att, const int* __restrict__ ei,
                             float* __restrict__ logits, float* __restrict__ nmax) {
  __shared__ float sW[EED * OUTC];         // 16 KB
  for (int i = threadIdx.x; i < EED * OUTC; i += blockDim.x) sW[i] = Wed[i];
  __syncthreads();
  int wid  = blockIdx.x * 8 + (threadIdx.x >> 5);
  int lane = threadIdx.x & 31;
  if (wid >= EE) return;
  int s = ei[wid], d = ei[EE + wid];
  const float* xld = xl + (size_t)d * OUTC;
  const float* xrs = xr + (size_t)s * OUTC;
  float ev[EED];
#pragma unroll
  for (int k = 0; k < EED; k++) ev[k] = e1[(size_t)wid * EED + k];
  float acc = 0.f;
#pragma unroll
  for (int jj = 0; jj < 8; jj++) {
    int c = jj * 32 + lane;
    float ep = 0.f;
#pragma unroll
    for (int k = 0; k < EED; k++) ep = fmaf(ev[k], sW[k * OUTC + c], ep);
    float m = xld[c] + xrs[c] + ep;
    m = m > 0.f ? m : 0.2f * m;
    acc = fmaf(att[c], m, acc);
  }
  acc = wave_sum(acc);
  if (lane == 0) {
    logits[wid] = acc;
    atomicMax(&nmax[d], acc);
  }
}

__global__ void edge_exp2(float* __restrict__ a, const float* __restrict__ nmax,
                          float* __restrict__ nden, const int* __restrict__ ei) {
  int e = blockIdx.x * blockDim.x + threadIdx.x;
  if (e >= EE) return;
  int d = ei[EE + e];
  float x = expf(a[e] - nmax[d]);
  a[e] = x;
  atomicAdd(&nden[d], x);
}

__global__ void edge_aggr2(const float* __restrict__ xr, const float* __restrict__ a,
                           const float* __restrict__ nden, const int* __restrict__ ei,
                           float* __restrict__ out) {
  int wid  = blockIdx.x * 8 + (threadIdx.x >> 5);
  int lane = threadIdx.x & 31;
  if (wid >= EE) return;
  int s = ei[wid], d = ei[EE + wid];
  float alpha = a[wid] / (nden[d] + 1e-16f);
  const float* xrs = xr + (size_t)s * OUTC;
  float* od = out + (size_t)d * OUTC;
#pragma unroll
  for (int jj = 0; jj < 8; jj++) {
    int c = jj * 32 + lane;
    atomicAdd(&od[c], xrs[c] * alpha);
  }
}

// layernorm(out2 + bias2) then fused segment mean/max pooling accumulation.
__global__ void ln2_pool(const float* __restrict__ out2, const float* __restrict__ bias,
                         const float* __restrict__ g, const float* __restrict__ beta,
                         const int* __restrict__ batch,
                         float* __restrict__ psum, float* __restrict__ pmax,
                         float* __restrict__ pcnt) {
  int n = blockIdx.x * 8 + (threadIdx.x >> 5);
  int lane = threadIdx.x & 31;
  if (n >= NN) return;
  const float* row = out2 + (size_t)n * OUTC;
  float v[8];
  float s = 0.f;
#pragma unroll
  for (int j = 0; j < 8; j++) { int c = j * 32 + lane; v[j] = row[c] + bias[c]; s += v[j]; }
  float mu = wave_sum(s) * (1.f / OUTC);
  float q = 0.f;
#pragma unroll
  for (int j = 0; j < 8; j++) { float dd = v[j] - mu; q += dd * dd; }
  float var = wave_sum(q) * (1.f / OUTC);
  float rs = rsqrtf(var + 1e-5f);
  int grp = batch[n];
  float* ps = psum + (size_t)grp * OUTC;
  float* pm = pmax + (size_t)grp * OUTC;
#pragma unroll
  for (int j = 0; j < 8; j++) {
    int c = j * 32 + lane;
    float h = (v[j] - mu) * rs * g[c] + beta[c];
    atomicAdd(&ps[c], h);
    atomicMax(&pm[c], h);
  }
  if (lane == 0) atomicAdd(&pcnt[grp], 1.f);
}

__global__ void pool_final(const float* __restrict__ psum, const float* __restrict__ pmax,
                           const float* __restrict__ pcnt, float* __restrict__ out) {
  int i = blockIdx.x * blockDim.x + threadIdx.x;
  if (i >= GG * OUTC) return;
  int grp = i >> 8;
  float c = pcnt[grp];
  c = c < 1.f ? 1.f : c;
  out[i] = psum[i] / c + pmax[i];
}

extern "C" void kernel_launch(void* const* d_in, const int* in_sizes, int n_in,
                              void* d_out, int out_size, void* d_ws, size_t ws_size,
                              hipStream_t stream) {
  (void)in_sizes; (void)n_in; (void)out_size; (void)ws_size;
  const float* x     = (const float*)d_in[0];
  const int*   ei    = (const int*)  d_in[1];   // [2,E]: src=ei[0:E], dst=ei[E:2E]
  const float* eattr = (const float*)d_in[2];
  const int*   batch = (const int*)  d_in[3];
  const float* We_a  = (const float*)d_in[4];
  const float* be_a  = (const float*)d_in[5];
  const float* We_b  = (const float*)d_in[6];
  const float* be_b  = (const float*)d_in[7];
  const float* Wl1   = (const float*)d_in[8];
  const float* bl1   = (const float*)d_in[9];
  const float* Wr1   = (const float*)d_in[10];
  const float* br1   = (const float*)d_in[11];
  const float* Wed1  = (const float*)d_in[12];
  const float* att1  = (const float*)d_in[13];
  const float* bias1 = (const float*)d_in[14];
  const float* g1    = (const float*)d_in[15];
  const float* beta1 = (const float*)d_in[16];
  const float* Wl2   = (const float*)d_in[17];
  const float* bl2   = (const float*)d_in[18];
  const float* Wr2   = (const float*)d_in[19];
  const float* br2   = (const float*)d_in[20];
  const float* Wed2  = (const float*)d_in[21];
  const float* att2  = (const float*)d_in[22];
  const float* bias2 = (const float*)d_in[23];
  const float* g2    = (const float*)d_in[24];
  const float* beta2 = (const float*)d_in[25];
  float* out = (float*)d_out;

  // ---- workspace layout (256B aligned slabs, ~178 MB total) ----
  char* ws = (char*)d_ws;
  size_t off = 0;
  auto nxt = [&](size_t bytes) -> void* {
    void* p = ws + off;
    off += (bytes + 255) & ~(size_t)255;
    return p;
  };
  float* e1    = (float*)nxt((size_t)EE * EED * 4);
  float* regA  = (float*)nxt((size_t)NN * C1 * 4);   // xl1 -> out1/h1
  float* regB  = (float*)nxt((size_t)NN * C1 * 4);   // xr1 -> xl2|xr2|out2
  float* log1  = (float*)nxt((size_t)EE * NH1 * 4);
  float* nmax1 = (float*)nxt((size_t)NN * NH1 * 4);
  float* nden1 = (float*)nxt((size_t)NN * NH1 * 4);
  float* log2  = (float*)nxt((size_t)EE * 4);
  float* nmax2 = (float*)nxt((size_t)NN * 4);
  float* nden2 = (float*)nxt((size_t)NN * 4);
  float* psum  = (float*)nxt((size_t)GG * OUTC * 4);
  float* pmax  = (float*)nxt((size_t)GG * OUTC * 4);
  float* pcnt  = (float*)nxt((size_t)GG * 4);
  float* xl1 = regA;
  float* xr1 = regB;
  float* h1  = regA;                                  // aliases out1 (zeroed post-logits)
  float* xl2 = regB;                                  // xr1 dead after aggr1
  float* xr2 = regB + (size_t)NN * OUTC;
  float* out2 = regB + (size_t)2 * NN * OUTC;

  const int T = 256;
  auto blks = [](long long n, int t) { return (int)((n + t - 1) / t); };

  // ---- init accumulators (every call; ws is not re-poisoned between replays) ----
  fill_f32<<<blks(NN * NH1, T), T, 0, stream>>>(nmax1, -FLT_MAX, NN * NH1);
  fill_f32<<<blks(NN * NH1, T), T, 0, stream>>>(nden1, 0.f, NN * NH1);
  fill_f32<<<blks(NN, T), T, 0, stream>>>(nmax2, -FLT_MAX, NN);
  fill_f32<<<blks(NN, T), T, 0, stream>>>(nden2, 0.f, NN);
  fill_f32<<<blks(GG * OUTC, T), T, 0, stream>>>(psum, 0.f, GG * OUTC);
  fill_f32<<<blks(GG * OUTC, T), T, 0, stream>>>(pmax, -FLT_MAX, GG * OUTC);
  fill_f32<<<blks(GG, T), T, 0, stream>>>(pcnt, 0.f, GG);

  // ---- edge encoder ----
  edge_encoder<<<blks(EE, T), T, 0, stream>>>(eattr, We_a, be_a, We_b, be_b, e1);

  // ---- conv1 dense GEMMs (WMMA f32): [20000,512]x[512,1024] ----
  {
    int waves = (NN / 16) * (C1 / 64);
    gemm_wmma<<<blks(waves, 8), T, 0, stream>>>(x, Wl1, bl1, xl1, NN, DIN, C1);
    gemm_wmma<<<blks(waves, 8), T, 0, stream>>>(x, Wr1, br1, xr1, NN, DIN, C1);
  }

  // ---- conv1 segment softmax + aggregation ----
  edge_logits1<<<blks(EE, 8), T, 0, stream>>>(xl1, xr1, e1, Wed1, att1, ei, log1, nmax1);
  edge_exp1<<<blks((long long)EE * NH1, T), T, 0, stream>>>(log1, nmax1, nden1, ei);
  fill_f32<<<blks((long long)NN * C1, T), T, 0, stream>>>(regA, 0.f, NN * C1);  // out1 := 0
  edge_aggr1<<<blks(EE, 8), T, 0, stream>>>(xr1, log1, nden1, ei, regA);

  // ---- bias + LN + GELU (in place) ----
  ln_gelu1<<<blks(NN, 8), T, 0, stream>>>(h1, bias1, g1, beta1);

  // ---- conv2 dense GEMMs (WMMA f32): [20000,1024]x[1024,256] ----
  {
    int waves = (NN / 16) * (OUTC / 64);
    gemm_wmma<<<blks(waves, 8), T, 0, stream>>>(h1, Wl2, bl2, xl2, NN, C1, OUTC);
    gemm_wmma<<<blks(waves, 8), T, 0, stream>>>(h1, Wr2, br2, xr2, NN, C1, OUTC);
  }

  // ---- conv2 segment softmax + aggregation ----
  edge_logits2<<<blks(EE, 8), T, 0, stream>>>(xl2, xr2, e1, Wed2, att2, ei, log2, nmax2);
  edge_exp2<<<blks(EE, T), T, 0, stream>>>(log2, nmax2, nden2, ei);
  fill_f32<<<blks((long long)NN * OUTC, T), T, 0, stream>>>(out2, 0.f, NN * OUTC);
  edge_aggr2<<<blks(EE, 8), T, 0, stream>>>(xr2, log2, nden2, ei, out2);

  // ---- bias + LN + fused segment mean/max pooling, then combine ----
  ln2_pool<<<blks(NN, 8), T, 0, stream>>>(out2, bias2, g2, beta2, batch, psum, pmax, pcnt);
  pool_final<<<blks(GG * OUTC, T), T, 0, stream>>>(psum, pmax, pcnt, out);
}